// ProposalSelectorLayer_77713138254448
// MI455X (gfx1250) — compile-verified
//
#include <hip/hip_runtime.h>
#include <stdint.h>

// ---------------------------------------------------------------------------
// ProposalSelector for MI455X (gfx1250).
// Phase 1 (decode): bandwidth-bound stream of 85MB cls_score (argmax over 81
//   classes) + box decode/clip/mask.  ~4us roofline at 23.3 TB/s.
// Phase 2 (NMS): persistent cooperative kernel, 128 blocks x 256 thr.  Each
//   block stages its 2048-proposal slice (boxes 32KB + scores 8KB) into LDS
//   via the Tensor Data Mover, then runs 100 fused suppress+argmax rounds
//   entirely out of LDS.  Cross-block argmax per round = one packed u64
//   atomic_fetch_max per block + ONE grid barrier (100 total round trips).
// ---------------------------------------------------------------------------

#define CLS_THRESHOLD 0.7f
#define NMS_THRESHOLD 0.5f
#define NMS_MAX_OUT   100
#define MIN_SIZE      3.0f
#define NEGV          (-1e30f)
#define TPB           256
#define SLICE         2048          // proposals per NMS block (48KB LDS)

typedef unsigned int u32x4 __attribute__((ext_vector_type(4)));
typedef int          i32x4 __attribute__((ext_vector_type(4)));
typedef int          i32x8 __attribute__((ext_vector_type(8)));

#if defined(__HIP_DEVICE_COMPILE__) && __has_builtin(__builtin_amdgcn_tensor_load_to_lds)
#define USE_TDM 1
#else
#define USE_TDM 0
#endif

#if USE_TDM
// 1-row tensor load: ndwords contiguous dwords from `g` into LDS at lds_byte.
// D# per CDNA5 ISA ch.8: group0 = {count=1, lds_addr, global_addr, type=2},
// group1 = {data_size=4B, tensor_dim0=ndwords, tensor_dim1=1,
//           tile_dim0=ndwords, tile_dim1=1, dim0_stride=ndwords}.
__device__ __forceinline__ void tdm_load_row(uint32_t lds_byte, const void* g,
                                             uint32_t ndwords) {
  uint64_t ga = (uint64_t)(uintptr_t)g;
  u32x4 g0;
  g0[0] = 1u;                                             // count=1, user mode
  g0[1] = lds_byte;                                       // lds_addr
  g0[2] = (uint32_t)ga;                                   // global_addr[31:0]
  g0[3] = (uint32_t)((ga >> 32) & 0x1FFFFFFu) | (2u << 30); // [56:32] | type=2
  i32x8 g1;
  g1[0] = 0x00020000;                                     // data_size = 4B
  g1[1] = (int)((ndwords & 0xFFFFu) << 16);               // tensor_dim0 lo
  g1[2] = (int)(((ndwords >> 16) & 0xFFFFu) | (1u << 16)); // dim0 hi | dim1=1
  g1[3] = (int)((ndwords & 0xFFFFu) << 16);               // tile_dim0
  g1[4] = 1;                                              // tile_dim1=1
  g1[5] = (int)ndwords;                                   // dim0_stride lo
  g1[6] = 0; g1[7] = 0;
  i32x4 g2 = {0, 0, 0, 0};
  i32x4 g3 = {0, 0, 0, 0};
#if __clang_major__ >= 23
  i32x8 g4 = {0, 0, 0, 0, 0, 0, 0, 0};
  __builtin_amdgcn_tensor_load_to_lds(g0, g1, g2, g3, g4, 0);
#else
  __builtin_amdgcn_tensor_load_to_lds(g0, g1, g2, g3, 0);
#endif
}
__device__ __forceinline__ void tdm_wait() {
#if __has_builtin(__builtin_amdgcn_s_wait_tensorcnt)
  __builtin_amdgcn_s_wait_tensorcnt(0);
#else
  asm volatile("s_wait_tensorcnt 0x0" ::: "memory");
#endif
}
#endif  // USE_TDM

#if defined(__HIP_DEVICE_COMPILE__)
#define SPIN_RELAX() __builtin_amdgcn_s_sleep(1)
#else
#define SPIN_RELAX()
#endif

// Order-preserving float -> uint key (total order, max-compatible).
__device__ __forceinline__ uint32_t fkey(float f) {
  uint32_t b = __float_as_uint(f);
  return (b & 0x80000000u) ? ~b : (b | 0x80000000u);
}
__device__ __forceinline__ float funkey(uint32_t k) {
  uint32_t b = (k & 0x80000000u) ? (k & 0x7FFFFFFFu) : ~k;
  return __uint_as_float(b);
}

// ---------------------------------------------------------------------------
// Phase 1: per-proposal class argmax + box decode/clip/mask.
// Also (re)initializes the NMS round slots + grid-barrier state every call.
// ---------------------------------------------------------------------------
__global__ void __launch_bounds__(TPB)
decode_kernel(const float* __restrict__ proposals, const float* __restrict__ deltas,
              const float* __restrict__ cls, const int* __restrict__ shape,
              int N, int C,
              float* __restrict__ boxes, float* __restrict__ smask,
              float* __restrict__ ms, int* __restrict__ clsw,
              unsigned long long* __restrict__ slots, int* __restrict__ bstate) {
  int i = blockIdx.x * blockDim.x + threadIdx.x;
  if (i == 0) { bstate[0] = 0; bstate[1] = 0; }
  if (i < NMS_MAX_OUT) slots[i] = 0ull;           // identity for fetch_max
  if (i >= N) return;

  const float* row = cls + (size_t)i * C;
  __builtin_prefetch(row + 32, 0, 0);             // -> global_prefetch
  __builtin_prefetch(row + 64, 0, 0);
  float best = row[0]; int bi = 0;
  for (int j = 1; j < C; ++j) { float v = row[j]; if (v > best) { best = v; bi = j; } }

  float4 p = ((const float4*)proposals)[i];
  float4 d = ((const float4*)deltas)[i];
  float d0 = tanhf(d.x), d1 = tanhf(d.y), d2 = tanhf(d.z), d3 = tanhf(d.w);
  float y2 = d0 * p.z + p.x;
  float x2 = d1 * p.w + p.y;
  float h2 = (d2 + 1.0f) * p.z;
  float w2 = (d3 + 1.0f) * p.w;
  float H = (float)shape[1], W = (float)shape[2];
  float by1 = fminf(fmaxf(y2, 0.0f), H);
  float bx1 = fminf(fmaxf(x2, 0.0f), W);
  float by2 = fminf(fmaxf(y2 + h2, 0.0f), H);
  float bx2 = fminf(fmaxf(x2 + w2, 0.0f), W);
  bool keep = (best > CLS_THRESHOLD) && ((by2 - by1) > MIN_SIZE) && ((bx2 - bx1) > MIN_SIZE);

  ((float4*)boxes)[i] = make_float4(by1, bx1, by2, bx2);
  ms[i]    = best;
  smask[i] = keep ? best : NEGV;
  clsw[i]  = bi;
}

// ---------------------------------------------------------------------------
// Grid-wide sense-reversing barrier (agent scope, L2-coherent atomics).
// ---------------------------------------------------------------------------
__device__ __forceinline__ void grid_sync(int* ctr, int* sns, int nblk, int* lsense) {
  __syncthreads();
  if (threadIdx.x == 0) {
    int s = (*lsense ^= 1);
    int prev = __hip_atomic_fetch_add(ctr, 1, __ATOMIC_ACQ_REL, __HIP_MEMORY_SCOPE_AGENT);
    if (prev == nblk - 1) {
      __hip_atomic_store(ctr, 0, __ATOMIC_RELAXED, __HIP_MEMORY_SCOPE_AGENT);
      __hip_atomic_store(sns, s, __ATOMIC_RELEASE, __HIP_MEMORY_SCOPE_AGENT);
    } else {
      while (__hip_atomic_load(sns, __ATOMIC_ACQUIRE, __HIP_MEMORY_SCOPE_AGENT) != s)
        SPIN_RELAX();
    }
  }
  __syncthreads();
}

// ---------------------------------------------------------------------------
// Phase 2: persistent cooperative NMS.  One grid barrier per round.
// ---------------------------------------------------------------------------
__global__ void __launch_bounds__(TPB, 1)
nms_kernel(const float* __restrict__ boxes, const float* __restrict__ smask,
           const float* __restrict__ ms, const int* __restrict__ clsw,
           unsigned long long* __restrict__ slots, int* __restrict__ bstate,
           float* __restrict__ out, int N, int nblk) {
  __shared__ float4 sb[SLICE];       // 32KB  slice boxes
  __shared__ float  ss[SLICE];       //  8KB  slice scores (mutated)
  __shared__ float  sa[SLICE];       //  8KB  slice areas
  __shared__ float  rv[TPB / 32];
  __shared__ int    ri[TPB / 32];
  __shared__ float  wsh[8];          // winner {y1,x1,y2,x2,area}

  const int tid  = threadIdx.x;
  const int base = blockIdx.x * SLICE;
  const int cnt  = min(SLICE, N - base);

  // ---- stage slice into LDS ----
  for (int e = tid; e < SLICE; e += TPB) {            // pad tail first
    if (base + e >= N) { sb[e] = make_float4(0, 0, 0, 0); ss[e] = NEGV; }
  }
  if (tid < 8) wsh[tid] = 0.0f;
  __syncthreads();
#if USE_TDM
  if (tid < 32) {                                     // one TDM issue per block
    tdm_load_row((uint32_t)(uintptr_t)sb, boxes + (size_t)base * 4, (uint32_t)cnt * 4);
    tdm_load_row((uint32_t)(uintptr_t)ss, smask + base, (uint32_t)cnt);
    tdm_wait();
  }
#else
  for (int e = tid; e < cnt; e += TPB) {
    sb[e] = ((const float4*)boxes)[base + e];
    ss[e] = smask[base + e];
  }
#endif
  __syncthreads();
  for (int e = tid; e < SLICE; e += TPB) {
    float4 b = sb[e];
    sa[e] = (b.z - b.x) * (b.w - b.y);
  }
  __syncthreads();

  int lsense = 0;
  const int per = SLICE / TPB;                         // 8 elements / thread

  for (int k = 0; k < NMS_MAX_OUT; ++k) {
    float4 wb = make_float4(wsh[0], wsh[1], wsh[2], wsh[3]);
    float  wa = wsh[4];

    // ---- fused suppress (winner k-1) + local argmax ----
    float bestv = -3.4e38f;
    int   beste = 0;
    int   e0    = tid * per;
#pragma unroll
    for (int q = 0; q < per; ++q) {
      int e = e0 + q;
      float4 b = sb[e];
      if (k > 0) {
        float yy1 = fmaxf(wb.x, b.x), xx1 = fmaxf(wb.y, b.y);
        float yy2 = fminf(wb.z, b.z), xx2 = fminf(wb.w, b.w);
        float inter = fmaxf(yy2 - yy1, 0.0f) * fmaxf(xx2 - xx1, 0.0f);
        float iou = inter / (wa + sa[e] - inter + 1e-9f);
        if (iou > NMS_THRESHOLD) ss[e] = NEGV;
      }
      float s = ss[e];
      if (s > bestv) { bestv = s; beste = e; }
    }
    int gi = base + beste;
    for (int off = 16; off; off >>= 1) {               // wave32 reduce
      float ov = __shfl_xor(bestv, off, 32);
      int   og = __shfl_xor(gi, off, 32);
      if (ov > bestv || (ov == bestv && og < gi)) { bestv = ov; gi = og; }
    }
    if ((tid & 31) == 0) { rv[tid >> 5] = bestv; ri[tid >> 5] = gi; }
    __syncthreads();
    if (tid == 0) {
      float bv = rv[0]; int bg = ri[0];
      for (int wv = 1; wv < TPB / 32; ++wv)
        if (rv[wv] > bv || (rv[wv] == bv && ri[wv] < bg)) { bv = rv[wv]; bg = ri[wv]; }
      // pack: score key (hi) | 0x7FFFFFFF - idx (lo) -> max picks best score,
      // ties resolve to smallest index (jnp.argmax first-max semantics).
      unsigned long long packed =
          ((unsigned long long)fkey(bv) << 32) |
          (unsigned long long)(0x7FFFFFFFu - (uint32_t)bg);
      __hip_atomic_fetch_max(&slots[k], packed, __ATOMIC_RELAXED,
                             __HIP_MEMORY_SCOPE_AGENT);
    }
    grid_sync(&bstate[0], &bstate[1], nblk, &lsense);

    // ---- every block reconstructs winner k; block 0 emits outputs ----
    if (tid == 0) {
      unsigned long long packed =
          __hip_atomic_load(&slots[k], __ATOMIC_RELAXED, __HIP_MEMORY_SCOPE_AGENT);
      int   wgi = (int)(0x7FFFFFFFu - (uint32_t)(packed & 0xFFFFFFFFu));
      float wsc = funkey((uint32_t)(packed >> 32));
      float4 b  = ((const float4*)boxes)[wgi];
      float area = (b.z - b.x) * (b.w - b.y);
      wsh[0] = b.x; wsh[1] = b.y; wsh[2] = b.z; wsh[3] = b.w; wsh[4] = area;
      if (blockIdx.x == 0) {
        bool valid = wsc > NEGV * 0.5f;
        out[4 * k + 0] = valid ? b.x : 0.0f;
        out[4 * k + 1] = valid ? b.y : 0.0f;
        out[4 * k + 2] = valid ? b.z : 0.0f;
        out[4 * k + 3] = valid ? b.w : 0.0f;
        out[4 * NMS_MAX_OUT + k] = valid ? ms[wgi] : 0.0f;
        out[5 * NMS_MAX_OUT + k] = valid ? (float)clsw[wgi] : -1.0f;
      }
    }
    __syncthreads();
  }
}

// ---------------------------------------------------------------------------
extern "C" void kernel_launch(void* const* d_in, const int* in_sizes, int n_in,
                              void* d_out, int out_size, void* d_ws, size_t ws_size,
                              hipStream_t stream) {
  const float* proposals = (const float*)d_in[0];
  const float* deltas    = (const float*)d_in[1];
  const float* cls       = (const float*)d_in[2];
  const int*   shape     = (const int*)d_in[3];
  float* out = (float*)d_out;

  const int N = in_sizes[0] / 4;           // 262144
  const int C = in_sizes[2] / N;           // 81

  // workspace layout (bytes):
  char* w = (char*)d_ws;
  float* boxes = (float*)w;                               // 16N
  float* smask = (float*)(w + 16ull * N);                 // 4N
  float* ms    = (float*)(w + 20ull * N);                 // 4N
  int*   clsw  = (int*)  (w + 24ull * N);                 // 4N
  unsigned long long* slots = (unsigned long long*)(w + 28ull * N); // 800B
  int*   bstate= (int*)  (w + 28ull * N + 1024);          // 8B

  const int nblk = (N + SLICE - 1) / SLICE;               // 128

  decode_kernel<<<(N + TPB - 1) / TPB, TPB, 0, stream>>>(
      proposals, deltas, cls, shape, N, C, boxes, smask, ms, clsw, slots, bstate);
  nms_kernel<<<nblk, TPB, 0, stream>>>(
      boxes, smask, ms, clsw, slots, bstate, out, N, nblk);
}